// PatchConv1LayerClassifier_8117488190077
// MI455X (gfx1250) — compile-verified
//
#include <hip/hip_runtime.h>
#include <hip/hip_bf16.h>

typedef float v2f __attribute__((ext_vector_type(2)));
typedef float v8f __attribute__((ext_vector_type(8)));

#define SLOPE 0.01f

// ---------------------------------------------------------------------------
// K1: structural degrees via float atomics
// ---------------------------------------------------------------------------
__global__ void k_degrees(const int* __restrict__ src, const int* __restrict__ dst,
                          float* __restrict__ degO, float* __restrict__ degI, int nE) {
    int e = blockIdx.x * blockDim.x + threadIdx.x;
    if (e < nE) {
        atomicAdd(&degO[src[e]], 1.0f);
        atomicAdd(&degI[dst[e]], 1.0f);
    }
}

// ---------------------------------------------------------------------------
// K2: h[N,64] = (node_feats * rsqrt(max(degO,1))) @ W_conv[32,64]
// WMMA f32 16x16x4. One wave per 16-row slab; 4 N-tiles x 8 K-steps.
// W_conv (8 KB) is staged into LDS with GLOBAL_LOAD_ASYNC_TO_LDS_B128
// (ASYNCcnt path, no VGPR round-trip), then B fragments stream via ds loads.
// ---------------------------------------------------------------------------
__global__ __launch_bounds__(256) void k_conv_gemm(const float* __restrict__ feats,
                                                   const float* __restrict__ degO,
                                                   const float* __restrict__ Wc,
                                                   float* __restrict__ h, int nRows) {
    __shared__ float Ws[32 * 64];

    // Async stage: 512 x B128 transfers (8192 B), 2 per thread.
    {
        const unsigned lds0 =
            (unsigned)(uintptr_t)(__attribute__((address_space(3))) float*)Ws;
        const int t = threadIdx.x;
        const float* g0 = Wc + (size_t)t * 4;
        const float* g1 = Wc + (size_t)(t + 256) * 4;
        asm volatile("global_load_async_to_lds_b128 %0, %1, off"
                     :: "v"(lds0 + t * 16), "v"(g0) : "memory");
        asm volatile("global_load_async_to_lds_b128 %0, %1, off"
                     :: "v"(lds0 + (t + 256) * 16), "v"(g1) : "memory");
        asm volatile("s_wait_asynccnt 0" ::: "memory");
    }
    __syncthreads();

    const int wave = threadIdx.x >> 5;            // 0..7
    const int lane = threadIdx.x & 31;
    const int rowTile = blockIdx.x * 8 + wave;    // 16-row tile index
    const int mbase = rowTile * 16;
    if (mbase >= nRows) return;                   // wave-uniform: EXEC all-1 for WMMA

    const int l15 = lane & 15;
    const int hi = lane >> 4;                     // 0 for lanes 0-15, 1 for 16-31
    const int khalf = hi * 2;

    int r = mbase + l15;
    if (r >= nRows) r = nRows - 1;                // clamp loads on ragged tail
    const float scale = rsqrtf(fmaxf(degO[r], 1.0f));

    // Preload the 8 A fragments for K=0..31 (each lane: 2 consecutive K values)
    v2f a[8];
#pragma unroll
    for (int ks = 0; ks < 8; ++ks) {
        const int k = ks * 4 + khalf;
        a[ks].x = feats[r * 32 + k] * scale;
        a[ks].y = feats[r * 32 + k + 1] * scale;
    }

    const bool fullTile = (mbase + 16 <= nRows);  // wave-uniform

#pragma unroll
    for (int nt = 0; nt < 4; ++nt) {
        const int nb = nt * 16 + l15;
        v8f c = {};
#pragma unroll
        for (int ks = 0; ks < 8; ++ks) {
            const int k = ks * 4 + khalf;
            v2f b;
            b.x = Ws[k * 64 + nb];
            b.y = Ws[(k + 1) * 64 + nb];
            c = __builtin_amdgcn_wmma_f32_16x16x4_f32(
                    false, a[ks], false, b, (short)0, c, false, false);
        }
        const int rbase = mbase + hi * 8;
        float* hp = h + (size_t)rbase * 64 + nb;
        if (fullTile) {
#pragma unroll
            for (int v = 0; v < 8; ++v) hp[v * 64] = c[v];
        } else {
#pragma unroll
            for (int v = 0; v < 8; ++v)
                if (rbase + v < nRows) hp[v * 64] = c[v];
        }
    }
}

// ---------------------------------------------------------------------------
// K3: edge gather-scale-scatter. 64 threads per edge (feature f = tid&63),
// coalesced 256B reads of h[src] and 256B of float atomics into agg[dst].
// ---------------------------------------------------------------------------
__global__ void k_edge_scatter(const float* __restrict__ h, const float* __restrict__ ew,
                               const int* __restrict__ src, const int* __restrict__ dst,
                               float* __restrict__ agg, int nE) {
    const int f = threadIdx.x & 63;
    const int e = blockIdx.x * (blockDim.x >> 6) + (threadIdx.x >> 6);
    if (e >= nE) return;
    const int s = src[e];
    const int d = dst[e];
    const float v = h[(size_t)s * 64 + f] * ew[e];
    atomicAdd(&agg[(size_t)d * 64 + f], v);
}

// ---------------------------------------------------------------------------
// K4: act = leaky_relu(agg * rsqrt(max(degI,1))) and per-graph pooling.
// graph_ids are sorted -> each thread accumulates a run of same-graph nodes
// and flushes one atomic per run (~32x fewer graph atomics).
// ---------------------------------------------------------------------------
#define CHUNK 32
__global__ void k_node_pool(const float* __restrict__ agg, const float* __restrict__ degI,
                            const int* __restrict__ gid, float* __restrict__ gsum,
                            float* __restrict__ gcnt, int nN) {
    const int f = threadIdx.x;                    // 0..63
    int node = blockIdx.x * CHUNK;
    const int nodeEnd = min(node + CHUNK, nN);
    if (node >= nN) return;

    int curG = gid[node];
    float acc = 0.0f;
    float cnt = 0.0f;
    for (; node < nodeEnd; ++node) {
        const int g = gid[node];
        if (g != curG) {
            atomicAdd(&gsum[(size_t)curG * 64 + f], acc);
            if (f == 0) atomicAdd(&gcnt[curG], cnt);
            curG = g; acc = 0.0f; cnt = 0.0f;
        }
        float v = agg[(size_t)node * 64 + f] * rsqrtf(fmaxf(degI[node], 1.0f));
        v = (v > 0.0f) ? v : v * SLOPE;
        acc += v;
        cnt += 1.0f;
    }
    atomicAdd(&gsum[(size_t)curG * 64 + f], acc);
    if (f == 0) atomicAdd(&gcnt[curG], cnt);
}

// ---------------------------------------------------------------------------
// K5: tiny per-graph MLP: pooled[64] -> t[32] (W1[32,64]) -> out[16] (W2[16,32])
// ---------------------------------------------------------------------------
__global__ void k_mlp(const float* __restrict__ gsum, const float* __restrict__ gcnt,
                      const float* __restrict__ W1, const float* __restrict__ W2,
                      float* __restrict__ out) {
    __shared__ float p[64];
    __shared__ float t[32];
    const int g = blockIdx.x;
    const int tid = threadIdx.x;
    const float inv = 1.0f / fmaxf(gcnt[g], 1.0f);
    p[tid] = gsum[(size_t)g * 64 + tid] * inv;
    __syncthreads();
    if (tid < 32) {
        float s = 0.0f;
#pragma unroll
        for (int k = 0; k < 64; ++k) s += p[k] * W1[tid * 64 + k];
        t[tid] = s;
    }
    __syncthreads();
    if (tid < 16) {
        float s = 0.0f;
#pragma unroll
        for (int j = 0; j < 32; ++j) s += t[j] * W2[tid * 32 + j];
        out[(size_t)g * 16 + tid] = s;
    }
}

// ---------------------------------------------------------------------------
extern "C" void kernel_launch(void* const* d_in, const int* in_sizes, int n_in,
                              void* d_out, int out_size, void* d_ws, size_t ws_size,
                              hipStream_t stream) {
    const float* node_feats = (const float*)d_in[0];
    const float* edge_w     = (const float*)d_in[1];
    const int*   edge_src   = (const int*)  d_in[2];
    const int*   edge_dst   = (const int*)  d_in[3];
    const int*   graph_ids  = (const int*)  d_in[4];
    const float* W_conv     = (const float*)d_in[5];
    const float* W1         = (const float*)d_in[6];
    const float* W2         = (const float*)d_in[7];
    float* out = (float*)d_out;

    const int N = in_sizes[0] / 32;
    const int E = in_sizes[1];
    const int G = out_size / 16;

    // Workspace layout: zero-initialized accumulators first, then h.
    float* ws   = (float*)d_ws;
    float* degO = ws;                              // N
    float* degI = degO + N;                        // N
    float* agg  = degI + N;                        // N*64
    float* gsum = agg + (size_t)N * 64;            // G*64
    float* gcnt = gsum + (size_t)G * 64;           // G (padded)
    float* h    = gcnt + ((G + 63) / 64) * 64;     // N*64
    const size_t zeroBytes = (size_t)((char*)h - (char*)ws);

    hipMemsetAsync(d_ws, 0, zeroBytes, stream);

    // K1: degrees
    k_degrees<<<(E + 255) / 256, 256, 0, stream>>>(edge_src, edge_dst, degO, degI, E);

    // K2: WMMA conv GEMM
    const int rowTiles = (N + 15) / 16;
    k_conv_gemm<<<(rowTiles + 7) / 8, 256, 0, stream>>>(node_feats, degO, W_conv, h, N);

    // K3: edge gather/scatter (4 edges per 256-thread block, 64 threads/edge)
    k_edge_scatter<<<(E + 3) / 4, 256, 0, stream>>>(h, edge_w, edge_src, edge_dst, agg, E);

    // K4: activation + graph mean-pool accumulation
    k_node_pool<<<(N + CHUNK - 1) / CHUNK, 64, 0, stream>>>(agg, degI, graph_ids, gsum, gcnt, N);

    // K5: per-graph MLP head
    k_mlp<<<G, 64, 0, stream>>>(gsum, gcnt, W1, W2, out);
}